// SensoryModule_52673478918338
// MI455X (gfx1250) — compile-verified
//
#include <hip/hip_runtime.h>
#include <cstdint>

typedef __attribute__((ext_vector_type(4))) float v4f;

#define NROWS 16384      // B
#define MCOLS 512        // M
#define C4    128        // M / 4  (float4 columns)
#define TILE_ROWS 8
#define THREADS 256

static __device__ __forceinline__ v4f relu4(v4f x) {
    v4f r;
    r.x = fmaxf(x.x, 0.0f);
    r.y = fmaxf(x.y, 0.0f);
    r.z = fmaxf(x.z, 0.0f);
    r.w = fmaxf(x.w, 0.0f);
    return r;
}

__global__ __launch_bounds__(THREADS)
void sensory_elemwise_kernel(const float* __restrict__ lengths,
                             const float* __restrict__ velocities,
                             const float* __restrict__ forces,
                             const float* __restrict__ gamma_static,
                             const float* __restrict__ gamma_dynamic,
                             const float* __restrict__ wIa, const float* __restrict__ bIa,
                             const float* __restrict__ wII, const float* __restrict__ bII,
                             const float* __restrict__ wIb, const float* __restrict__ bIb,
                             float* __restrict__ out)
{
    // 6 parameter vectors of M floats each = 12 KB staged in LDS.
    __shared__ float smem[6 * MCOLS];

    const int tid = threadIdx.x;

    // --- CDNA5 async global->LDS broadcast of the parameter vectors. ---
    // Threads 0..127 each copy one 16B chunk of each of the 6 vectors.
    if (tid < C4) {
        const float* srcs[6] = { wIa, bIa, wII, bII, wIb, bIb };
        const uint32_t voff = (uint32_t)(tid * 16);
#pragma unroll
        for (int j = 0; j < 6; ++j) {
            // Low 32 bits of the generic LDS pointer == group-segment byte offset.
            uint32_t lds_addr =
                (uint32_t)(uintptr_t)(&smem[j * MCOLS]) + voff;
            uint64_t gbase = (uint64_t)(uintptr_t)srcs[j];
            asm volatile("global_load_async_to_lds_b128 %0, %1, %2"
                         :
                         : "v"(lds_addr), "v"(voff), "s"(gbase)
                         : "memory");
        }
        asm volatile("s_wait_asynccnt 0" ::: "memory");
    }
    __syncthreads();

    // Thread -> (row half, float4 column) mapping within the 8-row tile.
    const int col = tid & (C4 - 1);   // 0..127
    const int rh  = tid >> 7;         // 0..1

    // Pull this thread's 4-muscle parameter slice from LDS into registers.
    const v4f wIa4 = *(const v4f*)&smem[0 * MCOLS + col * 4];
    const v4f bIa4 = *(const v4f*)&smem[1 * MCOLS + col * 4];
    const v4f wII4 = *(const v4f*)&smem[2 * MCOLS + col * 4];
    const v4f bII4 = *(const v4f*)&smem[3 * MCOLS + col * 4];
    const v4f wIb4 = *(const v4f*)&smem[4 * MCOLS + col * 4];
    const v4f bIb4 = *(const v4f*)&smem[5 * MCOLS + col * 4];

    const v4f* __restrict__ L  = (const v4f*)lengths;
    const v4f* __restrict__ V  = (const v4f*)velocities;
    const v4f* __restrict__ F  = (const v4f*)forces;
    const v4f* __restrict__ GS = (const v4f*)gamma_static;
    const v4f* __restrict__ GD = (const v4f*)gamma_dynamic;
    v4f* __restrict__ O = (v4f*)out;

    const int row0 = blockIdx.x * TILE_ROWS + rh * 4;

#pragma unroll
    for (int i = 0; i < 4; ++i) {
        const int row    = row0 + i;
        const int in_idx = row * C4 + col;

        // Single-use streams: non-temporal B128 loads.
        const v4f v  = __builtin_nontemporal_load(&V[in_idx]);
        const v4f gd = __builtin_nontemporal_load(&GD[in_idx]);
        const v4f l  = __builtin_nontemporal_load(&L[in_idx]);
        const v4f gs = __builtin_nontemporal_load(&GS[in_idx]);
        const v4f f  = __builtin_nontemporal_load(&F[in_idx]);

        const v4f tIa = relu4((v * gd) * wIa4 + bIa4);
        const v4f tII = relu4((l + gs) * wII4 + bII4);
        const v4f tIb = relu4(f * wIb4 + bIb4);

        // Output row layout: [Ia(0..M) | II(M..2M) | Ib(2M..3M)], 3*C4 float4s/row.
        const int ob = row * (3 * C4);
        __builtin_nontemporal_store(tIa, &O[ob + col]);
        __builtin_nontemporal_store(tII, &O[ob + C4 + col]);
        __builtin_nontemporal_store(tIb, &O[ob + 2 * C4 + col]);
    }
}

extern "C" void kernel_launch(void* const* d_in, const int* in_sizes, int n_in,
                              void* d_out, int out_size, void* d_ws, size_t ws_size,
                              hipStream_t stream) {
    const float* lengths       = (const float*)d_in[0];
    const float* velocities    = (const float*)d_in[1];
    const float* forces        = (const float*)d_in[2];
    const float* gamma_static  = (const float*)d_in[3];
    const float* gamma_dynamic = (const float*)d_in[4];
    const float* w_Ia          = (const float*)d_in[5];
    const float* b_Ia          = (const float*)d_in[6];
    const float* w_II          = (const float*)d_in[7];
    const float* b_II          = (const float*)d_in[8];
    const float* w_Ib          = (const float*)d_in[9];
    const float* b_Ib          = (const float*)d_in[10];
    float* out                 = (float*)d_out;

    dim3 grid(NROWS / TILE_ROWS);   // 2048 blocks
    dim3 block(THREADS);            // 8 wave32 per block
    sensory_elemwise_kernel<<<grid, block, 0, stream>>>(
        lengths, velocities, forces, gamma_static, gamma_dynamic,
        w_Ia, b_Ia, w_II, b_II, w_Ib, b_Ib, out);
}